// Max_Pooling_Layer_TF_59347858096189
// MI455X (gfx1250) — compile-verified
//
#include <hip/hip_runtime.h>

// 4D VALID max-pool, window=strides=(2,2,2,2) over (H,W,D,F) of
// x : (N=4, H=64, W=64, D=32, F=16, C=32) f32  ->  out : (4,32,32,16,8,32)
//
// Input element strides:  C:1  F:32  D:512  W:16384  H:1048576  N:67108864
// Output element strides: C:1  F:32  D:256  W:4096   H:131072   N:4194304
//
// Thread tid handles one float4 (4 channels) of one output point; the
// decomposition below makes the output address exactly ((float4*)out)[tid].

typedef float f4 __attribute__((ext_vector_type(4)));

__global__ __launch_bounds__(256) void
Max_Pooling_Layer_TF_59347858096189_kernel(const float* __restrict__ x,
                                           float* __restrict__ out,
                                           unsigned total_vec4) {
    const unsigned tid = blockIdx.x * blockDim.x + threadIdx.x;
    if (tid >= total_vec4) return;

    // tid = (((((n*32 + ho)*32 + wo)*16 + do_)*8 + fo)*8 + k)
    const unsigned k   =  tid        & 7u;   // channel quad: c = 4k
    const unsigned fo  = (tid >> 3)  & 7u;
    const unsigned d_o = (tid >> 6)  & 15u;
    const unsigned wo  = (tid >> 10) & 31u;
    const unsigned ho  = (tid >> 15) & 31u;
    const unsigned n   =  tid >> 20;

    // Input base for (2ho, 2wo, 2do, 2fo, 4k); all 16 taps are immediate
    // offsets from this single address (max 4,262,016 B < 2^23, fits IOFFSET).
    const size_t base = (size_t)n   * 67108864u   // 2^26 per batch
                      + (size_t)ho  * 2097152u    // 2 * H-stride
                      + (size_t)wo  * 32768u      // 2 * W-stride
                      + (size_t)d_o * 1024u       // 2 * D-stride
                      + (size_t)fo  * 64u         // 2 * F-stride
                      + (size_t)k   * 4u;
    const float* p = x + base;

    // Tap 0 initializes the accumulator; remaining 15 taps max-accumulate.
    f4 m = __builtin_nontemporal_load((const f4*)p);

#pragma unroll
    for (int dh = 0; dh < 2; ++dh)
#pragma unroll
        for (int dw = 0; dw < 2; ++dw)
#pragma unroll
            for (int dd = 0; dd < 2; ++dd)
#pragma unroll
                for (int df = 0; df < 2; ++df) {
                    if ((dh | dw | dd | df) == 0) continue;
                    const size_t off = (size_t)dh * 1048576u
                                     + (size_t)dw * 16384u
                                     + (size_t)dd * 512u
                                     + (size_t)df * 32u;
                    const f4 v = __builtin_nontemporal_load((const f4*)(p + off));
                    m = __builtin_elementwise_max(m, v);
                }

    __builtin_nontemporal_store(m, (f4*)out + tid);
}

extern "C" void kernel_launch(void* const* d_in, const int* in_sizes, int n_in,
                              void* d_out, int out_size, void* d_ws, size_t ws_size,
                              hipStream_t stream) {
    (void)in_sizes; (void)n_in; (void)d_ws; (void)ws_size;
    const float* x = (const float*)d_in[0];
    float* out = (float*)d_out;

    const unsigned total_vec4 = (unsigned)(out_size / 4);  // 4,194,304
    const unsigned block = 256;
    const unsigned grid = (total_vec4 + block - 1) / block; // 16,384

    Max_Pooling_Layer_TF_59347858096189_kernel<<<grid, block, 0, stream>>>(
        x, out, total_vec4);
}